// RGCN_48129403519564
// MI455X (gfx1250) — compile-verified
//
#include <hip/hip_runtime.h>
#include <cstdint>

#define N_NODES 50000
#define N_EDGES 800000
#define IN_F    64
#define HID_F   64
#define OUT_F   32
#define N_REL   8
#define N_BASES 8
#define BN_EPS  1e-5f

typedef __attribute__((ext_vector_type(2))) float v2f;
typedef __attribute__((ext_vector_type(8))) float v8f;

__device__ __forceinline__ void atomic_add_f32(float* p, float v) {
    // lowers to global_atomic_add_f32 (relaxed, device scope)
    __hip_atomic_fetch_add(p, v, __ATOMIC_RELAXED, __HIP_MEMORY_SCOPE_AGENT);
}

// ---------------------------------------------------------------------------
// Build Wcat[k][c] for c in [0, R*O): sum_b comp[r,b]*bases[b,k,o]  (c = r*O+o)
//                  for c in [R*O, R*O+O): root[k, c-R*O]
// ---------------------------------------------------------------------------
__global__ void combine_weights_kernel(const float* __restrict__ comp,
                                       const float* __restrict__ bases,
                                       const float* __restrict__ root,
                                       float* __restrict__ Wc, int K, int O) {
    int TOT = N_REL * O + O;
    int idx = blockIdx.x * blockDim.x + threadIdx.x;
    if (idx >= K * TOT) return;
    int k = idx / TOT;
    int c = idx % TOT;
    float v;
    if (c < N_REL * O) {
        int r = c / O, o = c % O;
        v = 0.f;
        #pragma unroll
        for (int b = 0; b < N_BASES; ++b)
            v += comp[r * N_BASES + b] * bases[((size_t)b * K + k) * O + o];
    } else {
        v = root[(size_t)k * O + (c - N_REL * O)];
    }
    Wc[idx] = v;
}

// ---------------------------------------------------------------------------
// WMMA GEMM:  [N x 64] * [64 x TOT] -> cols < R*O go to Y, last O cols (+bias)
// go to rootOut.  One wave = one 16x16 tile, 4 waves per block.
// B panel (64x64 cols of W) staged in LDS, padded stride 68 so the two
// lane-halves' ds_load_b64 reads hit disjoint bank sets.
// 16 chained v_wmma_f32_16x16x4_f32 per tile (K = 64).
// ---------------------------------------------------------------------------
template <int O, int TOT>
__global__ __launch_bounds__(128) void wmma_gemm_kernel(
    const float* __restrict__ X, const float* __restrict__ W,
    const float* __restrict__ bias, float* __restrict__ Y,
    float* __restrict__ rootOut) {
    constexpr int K = 64;
    constexpr int RO = TOT - O;
    constexpr int BSTRIDE = 68;                   // pad: conflict-free b64 reads

    __shared__ float Bsh[64 * BSTRIDE];           // 17 KB

    int wave = threadIdx.x >> 5;
    int lane = threadIdx.x & 31;
    int laneHalf = lane & 15;
    bool hi = lane >= 16;

    int colBase0 = blockIdx.x * 64;               // block's first output column
    int rowBase  = blockIdx.y * 16;

    // ---- cooperative fill: W[0:64][colBase0:colBase0+64] -> LDS (col-major)
    for (int idx = threadIdx.x; idx < 64 * 64; idx += 128) {
        int c = idx & 63;                          // local column
        int k = idx >> 6;                          // k row
        int gcol = colBase0 + c;
        float v = (gcol < TOT) ? W[k * TOT + gcol] : 0.f;
        Bsh[c * BSTRIDE + k] = v;
    }
    __syncthreads();

    // A: 16x4 f32: lanes 0-15 -> K = kb+{0,1}; lanes 16-31 -> K = kb+{2,3}
    const float* arow = X + (size_t)(rowBase + laneHalf) * K + (hi ? 2 : 0);
    // B fragment source in LDS (k-contiguous per column)
    const float* bptr = &Bsh[(wave * 16 + laneHalf) * BSTRIDE + (hi ? 2 : 0)];

    v8f acc = {};
    #pragma unroll
    for (int kb = 0; kb < K; kb += 4) {
        v2f a = *(const v2f*)(arow + kb);          // global_load_b64, imm offset
        v2f b = *(const v2f*)(bptr + kb);          // ds_load_b64, imm offset
        acc = __builtin_amdgcn_wmma_f32_16x16x4_f32(
            /*neg_a=*/false, a, /*neg_b=*/false, b,
            /*c_mod=*/(short)0, acc, /*reuse_a=*/false, /*reuse_b=*/false);
    }

    int col = colBase0 + wave * 16 + laneHalf;
    if (col >= TOT) return;                        // after WMMAs: safe
    #pragma unroll
    for (int g = 0; g < 8; ++g) {
        int row = rowBase + g + (hi ? 8 : 0);
        float v = acc[g];
        if (col < RO)
            Y[(size_t)row * RO + col] = v;
        else
            rootOut[(size_t)row * O + (col - RO)] = v + bias[col - RO];
    }
}

// ---------------------------------------------------------------------------
// Edge counts per (dst, relation) segment — shared by all three layers.
// ---------------------------------------------------------------------------
__global__ void count_kernel(const int* __restrict__ dst,
                             const int* __restrict__ et,
                             int* __restrict__ cnt) {
    int e = blockIdx.x * blockDim.x + threadIdx.x;
    if (e >= N_EDGES) return;
    atomicAdd(&cnt[dst[e] * N_REL + et[e]], 1);
}

// ---------------------------------------------------------------------------
// Scatter: out[dst] += Y[src, r] / cnt[dst,r].  float2 payloads; for O=32
// each wave covers 2 edges so all lanes work.
// ---------------------------------------------------------------------------
template <int O>
__global__ __launch_bounds__(256) void scatter_kernel(
    const int* __restrict__ src, const int* __restrict__ dst,
    const int* __restrict__ et, const int* __restrict__ cnt,
    const float* __restrict__ Y, float* __restrict__ out) {
    constexpr int RO = N_REL * O;
    constexpr int PAIRS = O / 2;                   // float2 per edge: 32 or 16
    constexpr int EPW = 32 / PAIRS;                // edges per wave: 1 or 2
    int wave = threadIdx.x >> 5;
    int lane = threadIdx.x & 31;
    int sub = lane / PAIRS;                        // edge slot within wave
    int j = lane % PAIRS;                          // pair index within edge
    long e = ((long)blockIdx.x * 8 + wave) * EPW + sub;
    if (e >= N_EDGES) return;
    int s = src[e], d = dst[e], r = et[e];
    int c = cnt[d * N_REL + r];
    float inv = 1.0f / (float)(c > 1 ? c : 1);
    float2 y = ((const float2*)(Y + (size_t)s * RO + r * O))[j];
    float* op = out + (size_t)d * O + 2 * j;
    atomic_add_f32(op + 0, y.x * inv);
    atomic_add_f32(op + 1, y.y * inv);
}

// ---------------------------------------------------------------------------
// BN statistics over relu(pre): per-feature sum and sum-of-squares.
// ---------------------------------------------------------------------------
__global__ __launch_bounds__(256) void bn_stats_kernel(
    const float* __restrict__ pre, float* __restrict__ stats) {
    int f = threadIdx.x & 63;
    int rowsPerIter = gridDim.x * (blockDim.x >> 6);
    int row = blockIdx.x * (blockDim.x >> 6) + (threadIdx.x >> 6);
    float s = 0.f, s2 = 0.f;
    for (int n = row; n < N_NODES; n += rowsPerIter) {
        float v = pre[(size_t)n * HID_F + f];
        v = fmaxf(v, 0.f);
        s += v;
        s2 += v * v;
    }
    atomic_add_f32(&stats[f], s);
    atomic_add_f32(&stats[HID_F + f], s2);
}

// ---------------------------------------------------------------------------
// BN apply: out = (relu(pre)-mu)*rsqrt(var+eps)*gamma + beta (+ residual)
// ---------------------------------------------------------------------------
__global__ void bn_apply_kernel(const float* __restrict__ pre,
                                const float* __restrict__ stats,
                                const float* __restrict__ gamma,
                                const float* __restrict__ beta,
                                const float* __restrict__ resid,
                                float* __restrict__ out, int total) {
    int idx = blockIdx.x * blockDim.x + threadIdx.x;
    if (idx >= total) return;
    int f = idx & 63;
    const float invN = 1.0f / (float)N_NODES;
    float v = fmaxf(pre[idx], 0.f);
    float mu = stats[f] * invN;
    float var = stats[HID_F + f] * invN - mu * mu;
    float y = (v - mu) * rsqrtf(var + BN_EPS) * gamma[f] + beta[f];
    if (resid) y += resid[idx];
    out[idx] = y;
}

// ---------------------------------------------------------------------------
extern "C" void kernel_launch(void* const* d_in, const int* in_sizes, int n_in,
                              void* d_out, int out_size, void* d_ws, size_t ws_size,
                              hipStream_t stream) {
    const float* x      = (const float*)d_in[0];
    const int*   ei     = (const int*)d_in[1];
    const int*   et     = (const int*)d_in[2];
    const float* comp1  = (const float*)d_in[3];
    const float* bases1 = (const float*)d_in[4];
    const float* root1  = (const float*)d_in[5];
    const float* bias1  = (const float*)d_in[6];
    const float* comp2  = (const float*)d_in[7];
    const float* bases2 = (const float*)d_in[8];
    const float* root2  = (const float*)d_in[9];
    const float* bias2  = (const float*)d_in[10];
    const float* comp3  = (const float*)d_in[11];
    const float* bases3 = (const float*)d_in[12];
    const float* root3  = (const float*)d_in[13];
    const float* bias3  = (const float*)d_in[14];
    const float* gamma1 = (const float*)d_in[15];
    const float* beta1  = (const float*)d_in[16];
    const float* gamma2 = (const float*)d_in[17];
    const float* beta2  = (const float*)d_in[18];
    float* out = (float*)d_out;

    const int* src = ei;
    const int* dst = ei + N_EDGES;

    // ---- workspace carve (~143 MB; Y fits the 192 MB L2) ----
    char* p = (char*)d_ws;
    float* Y     = (float*)p; p += (size_t)N_NODES * (N_REL * HID_F) * sizeof(float); // 102.4 MB
    float* agg   = (float*)p; p += (size_t)N_NODES * HID_F * sizeof(float);
    float* h1    = (float*)p; p += (size_t)N_NODES * HID_F * sizeof(float);
    float* h2    = (float*)p; p += (size_t)N_NODES * HID_F * sizeof(float);
    int*   cnt   = (int*)p;   p += (size_t)N_NODES * N_REL * sizeof(int);
    float* Wc    = (float*)p; p += (size_t)IN_F * (N_REL * HID_F + HID_F) * sizeof(float);
    float* stats = (float*)p; p += 2 * HID_F * sizeof(float);

    // ---- edge counts (shared by all layers) ----
    hipMemsetAsync(cnt, 0, (size_t)N_NODES * N_REL * sizeof(int), stream);
    count_kernel<<<(N_EDGES + 255) / 256, 256, 0, stream>>>(dst, et, cnt);

    constexpr int TOT1 = N_REL * HID_F + HID_F;  // 576
    constexpr int TOT3 = N_REL * OUT_F + OUT_F;  // 288
    dim3 gemm1((TOT1 + 63) / 64, N_NODES / 16);  // 9 x 3125
    dim3 gemm3((TOT3 + 63) / 64, N_NODES / 16);  // 5 x 3125
    int scatterBlocks1 = (N_EDGES + 7) / 8;       // 1 edge / wave (O=64)
    int scatterBlocks3 = (N_EDGES + 15) / 16;     // 2 edges / wave (O=32)
    int bnBlocks = (N_NODES * HID_F + 255) / 256;

    // ---- layer 1: x -> h1 ----
    combine_weights_kernel<<<(IN_F * TOT1 + 255) / 256, 256, 0, stream>>>(
        comp1, bases1, root1, Wc, IN_F, HID_F);
    wmma_gemm_kernel<HID_F, TOT1><<<gemm1, 128, 0, stream>>>(x, Wc, bias1, Y, agg);
    scatter_kernel<HID_F><<<scatterBlocks1, 256, 0, stream>>>(src, dst, et, cnt, Y, agg);
    hipMemsetAsync(stats, 0, 2 * HID_F * sizeof(float), stream);
    bn_stats_kernel<<<256, 256, 0, stream>>>(agg, stats);
    bn_apply_kernel<<<bnBlocks, 256, 0, stream>>>(agg, stats, gamma1, beta1,
                                                  nullptr, h1, N_NODES * HID_F);

    // ---- layer 2: h1 -> h2 (with residual) ----
    combine_weights_kernel<<<(HID_F * TOT1 + 255) / 256, 256, 0, stream>>>(
        comp2, bases2, root2, Wc, HID_F, HID_F);
    wmma_gemm_kernel<HID_F, TOT1><<<gemm1, 128, 0, stream>>>(h1, Wc, bias2, Y, agg);
    scatter_kernel<HID_F><<<scatterBlocks1, 256, 0, stream>>>(src, dst, et, cnt, Y, agg);
    hipMemsetAsync(stats, 0, 2 * HID_F * sizeof(float), stream);
    bn_stats_kernel<<<256, 256, 0, stream>>>(agg, stats);
    bn_apply_kernel<<<bnBlocks, 256, 0, stream>>>(agg, stats, gamma2, beta2,
                                                  h1, h2, N_NODES * HID_F);

    // ---- layer 3: h2 -> out (no relu/BN) ----
    combine_weights_kernel<<<(HID_F * TOT3 + 255) / 256, 256, 0, stream>>>(
        comp3, bases3, root3, Wc, HID_F, OUT_F);
    wmma_gemm_kernel<OUT_F, TOT3><<<gemm3, 128, 0, stream>>>(h2, Wc, bias3, Y, out);
    scatter_kernel<OUT_F><<<scatterBlocks3, 256, 0, stream>>>(src, dst, et, cnt, Y, out);
}